// GPRGNN_21784074125532
// MI455X (gfx1250) — compile-verified
//
#include <hip/hip_runtime.h>

typedef __attribute__((ext_vector_type(2))) float v2f;
typedef __attribute__((ext_vector_type(8))) float v8f;

#define PANEL 128   // K-panel rows staged per round (128*128*4B = 64 KB LDS)
#define NH    128   // hidden width (N_HID in the reference)

// ---- CDNA5 async copy: global -> LDS, 16B per lane, tracked by ASYNCcnt ----
__device__ __forceinline__ void async_b128(unsigned lds_off, const float* g) {
  asm volatile("global_load_async_to_lds_b128 %0, %1, off"
               :: "v"(lds_off), "v"(g) : "memory");
}
__device__ __forceinline__ void wait_async0() {
  asm volatile("s_wait_asynccnt 0x0" ::: "memory");
}

// ---------------------------------------------------------------- degree / norm
__global__ void gpr_init_deg(float* deg, int n) {
  int i = blockIdx.x * blockDim.x + threadIdx.x;
  if (i < n) deg[i] = 1.0f;  // self-loop weight
}

__global__ void gpr_deg_scatter(const int* __restrict__ col, float* deg, int nE) {
  int e = blockIdx.x * blockDim.x + threadIdx.x;
  if (e < nE) atomicAdd(&deg[col[e]], 1.0f);
}

__global__ void gpr_dinv(const float* __restrict__ deg, float* dinv, int n) {
  int i = blockIdx.x * blockDim.x + threadIdx.x;
  if (i < n) {
    float d = deg[i];
    dinv[i] = d > 0.0f ? rsqrtf(d) : 0.0f;
  }
}

__global__ void gpr_norm(const int* __restrict__ row, const int* __restrict__ col,
                         const float* __restrict__ dinv, float* nrm, int nE) {
  int e = blockIdx.x * blockDim.x + threadIdx.x;
  if (e < nE) nrm[e] = dinv[row[e]] * dinv[col[e]];
}

// ------------------------------------------------- f32 WMMA GEMM: D = A*B + bias
// N fixed at 128: 8 waves/block, wave w owns n-tile w. B is async-staged into LDS
// in 128-row panels shared by all 8 waves; A streams from global (WGP$-cached).
// V_WMMA_F32_16X16X4_F32 keeps the MLP bit-exact f32.
// A 16x4 frag: lane(lid)=M; VGPR0/1 = K+0/K+1 (lanes 0-15), K+2/K+3 (lanes 16-31).
// B 4x16 frag: VGPR0/1 = rows K+0/K+1 (lanes 0-15 = N), rows K+2/K+3 (16-31).
// C/D: VGPR j = M=j (lanes 0-15), M=j+8 (lanes 16-31), N = lane&15.
__global__ void gpr_wmma_gemm(const float* __restrict__ A, const float* __restrict__ B,
                              const float* __restrict__ bias, float* __restrict__ C,
                              int M, int K, int applyRelu) {
  __shared__ float sB[PANEL * NH];
  const int lane = threadIdx.x & 31;
  const int wave = threadIdx.x >> 5;   // 0..7 -> n-tile
  const int half = lane >> 4;
  const int lid  = lane & 15;
  const int col0 = wave << 4;
  const int mTiles = (M + 15) >> 4;
  const unsigned sBoff = (unsigned)(uintptr_t)sB;  // generic low 32 bits == LDS offset

  for (int mt = blockIdx.x; mt < mTiles; mt += gridDim.x) {
    const int row0 = mt << 4;
    v8f acc = {};
    const float* ap = A + (size_t)(row0 + lid) * K + 2 * half;

    for (int kp = 0; kp < K; kp += PANEL) {
      // ---- async-stage B[kp..kp+127][0..127] into LDS ----
      const float* Bp = B + (size_t)kp * NH;
      for (int i = threadIdx.x; i < (PANEL * NH) / 4; i += blockDim.x)
        async_b128(sBoff + (unsigned)i * 16u, Bp + (size_t)i * 4);
      wait_async0();
      __syncthreads();

      // ---- 32 WMMAs over the panel (A: global b64, B: LDS ds reads) ----
      for (int kk = 0; kk < PANEL; kk += 4) {
        v2f a, b;
        a.x = ap[0];  a.y = ap[1];  ap += 4;             // A[m][kp+kk+2h .. +1]
        const float* bq = sB + (kk + 2 * half) * NH + col0 + lid;
        b.x = bq[0];  b.y = bq[NH];                      // B rows kk+2h, kk+2h+1
        acc = __builtin_amdgcn_wmma_f32_16x16x4_f32(false, a, false, b,
                                                    (short)0, acc, false, false);
      }
      __syncthreads();  // protect sB before next panel staging
    }

    const float bv = bias[col0 + lid];
#pragma unroll
    for (int j = 0; j < 8; ++j) {
      float v = acc[j] + bv;
      if (applyRelu) v = fmaxf(v, 0.0f);
      C[(size_t)(row0 + j + 8 * half) * NH + col0 + lid] = v;
    }
  }
}

// ---------------------------------------------------------------- propagation
__global__ void gpr_init_acc(float* __restrict__ out, const float* __restrict__ h,
                             const float* __restrict__ temp, size_t n) {
  size_t i = (size_t)blockIdx.x * blockDim.x + threadIdx.x;
  if (i < n) out[i] = temp[0] * h[i];
}

// next = dinv[i]^2 * hk   (self-loop edge of the normalized adjacency)
__global__ void gpr_selfloop(const float* __restrict__ dinv, const float* __restrict__ hk,
                             float* __restrict__ nx, int nNodes) {
  size_t n = (size_t)nNodes * NH;
  size_t i = (size_t)blockIdx.x * blockDim.x + threadIdx.x;
  if (i < n) {
    float d = dinv[i / NH];
    nx[i] = d * d * hk[i];
  }
}

// Wave processes edges in batches of 32: each lane loads one edge's (row,col,norm)
// coalesced, then __shfl (ds_bpermute, wave32) broadcasts edge j while the 32
// lanes x float4 gather hk[row] (L2-resident) and scatter-add into nx[col] with
// global_atomic_add_f32.
__global__ void gpr_edge_scatter(const int* __restrict__ row, const int* __restrict__ col,
                                 const float* __restrict__ nrm, const float* __restrict__ hk,
                                 float* __restrict__ nx, int nE) {
  const int lane = threadIdx.x & 31;
  const int gw = (int)((blockIdx.x * blockDim.x + threadIdx.x) >> 5);
  const int nW = (int)((gridDim.x * blockDim.x) >> 5);
  for (int base = gw * 32; base < nE; base += nW * 32) {
    const int e  = base + lane;
    const bool ok = e < nE;
    const int   rv = ok ? row[e] : 0;
    const int   cv = ok ? col[e] : 0;
    const float wv = ok ? nrm[e] : 0.0f;
    const int cnt = min(32, nE - base);
    for (int j = 0; j < cnt; ++j) {
      const int   r = __shfl(rv, j);
      const int   c = __shfl(cv, j);
      const float w = __shfl(wv, j);
      const float4 v = *(const float4*)(hk + (size_t)r * NH + lane * 4);
      float* dst = nx + (size_t)c * NH + lane * 4;
      atomicAdd(dst + 0, w * v.x);
      atomicAdd(dst + 1, w * v.y);
      atomicAdd(dst + 2, w * v.z);
      atomicAdd(dst + 3, w * v.w);
    }
  }
}

__global__ void gpr_accum(float* __restrict__ out, const float* __restrict__ nx,
                          const float* __restrict__ temp, int k, size_t n) {
  size_t i = (size_t)blockIdx.x * blockDim.x + threadIdx.x;
  if (i < n) out[i] += temp[k] * nx[i];
}

// ---------------------------------------------------------------------- driver
extern "C" void kernel_launch(void* const* d_in, const int* in_sizes, int n_in,
                              void* d_out, int out_size, void* d_ws, size_t ws_size,
                              hipStream_t stream) {
  const float* x    = (const float*)d_in[0];
  const int*   edge = (const int*)  d_in[1];
  const float* W1   = (const float*)d_in[2];
  const float* b1   = (const float*)d_in[3];
  const float* W2   = (const float*)d_in[4];
  const float* b2   = (const float*)d_in[5];
  const float* temp = (const float*)d_in[6];

  const int nHid   = in_sizes[3];            // 128 (== NH)
  const int nFeat  = in_sizes[2] / nHid;     // 512
  const int nNodes = in_sizes[0] / nFeat;    // 100000
  const int nEdges = in_sizes[1] / 2;        // 3200000
  const int Ksteps = in_sizes[6] - 1;        // 10

  const int* rowIdx = edge;                  // edge_index[0]
  const int* colIdx = edge + nEdges;         // edge_index[1]

  const size_t hElems = (size_t)nNodes * nHid;
  float* hkA  = (float*)d_ws;        // GEMM2 output / hk ping
  float* hkB  = hkA + hElems;        // GEMM1 output / hk pong
  float* deg  = hkB + hElems;
  float* dinv = deg + nNodes;
  float* nrm  = dinv + nNodes;
  float* out  = (float*)d_out;

  const int blkN = (nNodes + 255) / 256;
  const int blkE = (nEdges + 255) / 256;
  const int blkH = (int)((hElems + 255) / 256);

  // ---- GCN normalization
  gpr_init_deg   <<<blkN, 256, 0, stream>>>(deg, nNodes);
  gpr_deg_scatter<<<blkE, 256, 0, stream>>>(colIdx, deg, nEdges);
  gpr_dinv       <<<blkN, 256, 0, stream>>>(deg, dinv, nNodes);
  gpr_norm       <<<blkE, 256, 0, stream>>>(rowIdx, colIdx, dinv, nrm, nEdges);

  // ---- MLP: hkB = relu(x@W1+b1);  hkA = hkB@W2+b2   (blockDim = (NH/16)*32)
  {
    const int mTiles = (nNodes + 15) / 16;
    const int blocks = mTiles < 2048 ? mTiles : 2048;
    gpr_wmma_gemm<<<blocks, 256, 0, stream>>>(x,   W1, b1, hkB, nNodes, nFeat, 1);
    gpr_wmma_gemm<<<blocks, 256, 0, stream>>>(hkB, W2, b2, hkA, nNodes, nHid, 0);
  }

  // ---- acc = temp[0]*h ; K propagation rounds
  gpr_init_acc<<<blkH, 256, 0, stream>>>(out, hkA, temp, hElems);

  float* hk = hkA;
  float* nx = hkB;
  for (int k = 1; k <= Ksteps; ++k) {
    gpr_selfloop    <<<blkH, 256, 0, stream>>>(dinv, hk, nx, nNodes);
    gpr_edge_scatter<<<4096, 256, 0, stream>>>(rowIdx, colIdx, nrm, hk, nx, nEdges);
    gpr_accum       <<<blkH, 256, 0, stream>>>(out, nx, temp, k, hElems);
    float* t = hk; hk = nx; nx = t;
  }
}